// EquivLayerNorm_37254546325938
// MI455X (gfx1250) — compile-verified
//
#include <hip/hip_runtime.h>

#define SDIM    128
#define VDIM    128
#define NGRAPH  512
#define EPS_F   1e-6f
#define WPB     8            // waves per block (256 threads)

typedef float v4f __attribute__((ext_vector_type(4)));

__device__ __forceinline__ float wave_sum32(float x) {
#pragma unroll
    for (int m = 16; m >= 1; m >>= 1) x += __shfl_xor(x, m, 32);
    return x;
}

// Issue 3 async b128 loads: one node's v-row (1536B) -> LDS buffer.
// Per CDNA5 ISA the instruction offset is added to BOTH the
// global and the LDS address, so one (lds,addr) pair + offsets 0/512/1024 covers it.
__device__ __forceinline__ void async_stage_v(const float* v, int n,
                                              float* buf, int lane) {
    unsigned long long ga = (unsigned long long)(const void*)(v + (size_t)n * (3 * VDIM))
                          + (unsigned long long)(lane * 16);
    // generic pointer to LDS: low 32 bits are the wave-relative LDS offset
    unsigned lba = (unsigned)(size_t)((const char*)buf + lane * 16);
    asm volatile("global_load_async_to_lds_b128 %0, %1, off"
                 :: "v"(lba), "v"(ga) : "memory");
    asm volatile("global_load_async_to_lds_b128 %0, %1, off offset:512"
                 :: "v"(lba), "v"(ga) : "memory");
    asm volatile("global_load_async_to_lds_b128 %0, %1, off offset:1024"
                 :: "v"(lba), "v"(ga) : "memory");
}

__device__ __forceinline__ int lower_bound_ll(const long long* __restrict__ a,
                                              int n, long long key) {
    int lo = 0, hi = n;
    while (lo < hi) {
        int mid = (lo + hi) >> 1;
        if (a[mid] < key) lo = mid + 1; else hi = mid;
    }
    return lo;
}

// One workgroup per graph. Phase A: streaming reduce (RT hints -> lines land in
// L2). Tiny LDS reduction computes {mu, 1/var, 1/vmean}. Phase B: re-read the
// same ~800KB (L2 hits) with NT loads and write outputs with NT stores so the
// output stream does not evict other graphs' phase-A data from L2.
__global__ void __launch_bounds__(256) eln_fused(
    const float* __restrict__ s, const float* __restrict__ v,
    const long long* __restrict__ index,
    const float* __restrict__ weight, const float* __restrict__ bias,
    float* __restrict__ outs, float* __restrict__ outv, int N)
{
    __shared__ float stage[WPB * 2 * (3 * VDIM)];   // 8 waves * double buffer * 1536B
    __shared__ float s_part[WPB][4];
    __shared__ float s_par[4];
    __shared__ int   s_rng[2];

    const int g    = blockIdx.x;
    const int wib  = threadIdx.x >> 5;
    const int lane = threadIdx.x & 31;

    if (threadIdx.x == 0) {
        s_rng[0] = lower_bound_ll(index, N, (long long)g);
        s_rng[1] = lower_bound_ll(index, N, (long long)g + 1);
    }
    __syncthreads();
    const int start = s_rng[0], end = s_rng[1];

    float* wbuf = stage + wib * 2 * (3 * VDIM);     // this wave's [2][384] floats
    const int first = start + wib;

    // ---------------- Phase A: per-graph statistics ----------------
    float a_s1 = 0.f, a_s2 = 0.f, a_v2 = 0.f;
    if (first < end) async_stage_v(v, first, wbuf, lane);   // prime buffer 0
    int it = 0;
    for (int n = first; n < end; n += WPB, ++it) {
        float* curbuf = wbuf + (it & 1) * (3 * VDIM);
        if (n + WPB < end) {
            async_stage_v(v, n + WPB, wbuf + ((it + 1) & 1) * (3 * VDIM), lane);
            asm volatile("s_wait_asynccnt 0x3" ::: "memory");  // current node's 3 done
        } else {
            asm volatile("s_wait_asynccnt 0x0" ::: "memory");
        }

        // scalar branch: direct coalesced stream (float4/lane = 512B/wave/node)
        v4f sv = ((const v4f*)(s + (size_t)n * SDIM))[lane];
        a_s1 += sv.x + sv.y + sv.z + sv.w;
        v4f s2 = sv * sv;
        a_s2 += s2.x + s2.y + s2.z + s2.w;

        // vector branch: consume from LDS staging buffer
        const v4f* vb = (const v4f*)curbuf + lane;
        v4f q0 = vb[0] * vb[0], q1 = vb[32] * vb[32], q2 = vb[64] * vb[64];
        a_v2 += q0.x + q0.y + q0.z + q0.w
              + q1.x + q1.y + q1.z + q1.w
              + q2.x + q2.y + q2.z + q2.w;
    }
    a_s1 = wave_sum32(a_s1);
    a_s2 = wave_sum32(a_s2);
    a_v2 = wave_sum32(a_v2);
    if (lane == 0) {
        s_part[wib][0] = a_s1; s_part[wib][1] = a_s2; s_part[wib][2] = a_v2;
    }
    __syncthreads();

    if (threadIdx.x == 0) {
        float t1 = 0.f, t2 = 0.f, t3 = 0.f;
#pragma unroll
        for (int i = 0; i < WPB; ++i) { t1 += s_part[i][0]; t2 += s_part[i][1]; t3 += s_part[i][2]; }
        float cnt = fmaxf((float)(end - start), 1.0f);
        float mu  = t1 * (1.0f / SDIM) / cnt;
        float m2  = t2 * (1.0f / SDIM) / cnt;
        float var = fmaxf(m2 - mu * mu, EPS_F);     // E[(s-mu)^2] == E[s^2]-mu^2
        float vm  = fmaxf(t3 * (1.0f / VDIM) / cnt, EPS_F);
        s_par[0] = mu; s_par[1] = 1.0f / var; s_par[2] = 1.0f / vm;
    }
    __syncthreads();

    // ---------------- Phase B: apply (NT reads, NT writes) ----------------
    const float mu  = s_par[0];
    const float iv  = s_par[1];
    const float ivm = s_par[2];
    const v4f w = ((const v4f*)weight)[lane];       // feature index is node-invariant
    const v4f b = ((const v4f*)bias)[lane];

    for (int n = first; n < end; n += WPB) {
        const v4f* sp = (const v4f*)(s + (size_t)n * SDIM) + lane;
        v4f sv = __builtin_nontemporal_load(sp);
        v4f o  = (sv - mu) * iv * w + b;
        __builtin_nontemporal_store(o, (v4f*)(outs + (size_t)n * SDIM) + lane);

        const v4f* vp = (const v4f*)(v + (size_t)n * (3 * VDIM)) + lane;
        v4f* op = (v4f*)(outv + (size_t)n * (3 * VDIM)) + lane;
        v4f q0 = __builtin_nontemporal_load(vp);
        v4f q1 = __builtin_nontemporal_load(vp + 32);
        v4f q2 = __builtin_nontemporal_load(vp + 64);
        __builtin_nontemporal_store(q0 * ivm, op);
        __builtin_nontemporal_store(q1 * ivm, op + 32);
        __builtin_nontemporal_store(q2 * ivm, op + 64);
    }
}

extern "C" void kernel_launch(void* const* d_in, const int* in_sizes, int n_in,
                              void* d_out, int out_size, void* d_ws, size_t ws_size,
                              hipStream_t stream) {
    const float*     s      = (const float*)d_in[0];
    const float*     v      = (const float*)d_in[1];
    const long long* index  = (const long long*)d_in[2];   // int64 per reference
    const float*     weight = (const float*)d_in[3];
    const float*     bias   = (const float*)d_in[4];
    const int N = in_sizes[2];

    float* outs = (float*)d_out;
    float* outv = outs + (size_t)N * SDIM;

    eln_fused<<<NGRAPH, 32 * WPB, 0, stream>>>(s, v, index, weight, bias, outs, outv, N);
}